// DimMasking_20968030339392
// MI455X (gfx1250) — compile-verified
//
#include <hip/hip_runtime.h>

namespace {
constexpr int   kD       = 640;
constexpr int   kHid     = 64;
constexpr int   kIter    = 64;           // D * 0.1
constexpr float kInvTemp = 1.0f / 0.07f;
constexpr float kEps     = 1e-7f;
constexpr int   kRows    = 16;           // rows per workgroup (WMMA M)
constexpr int   kThreads = 256;          // 8 wave32 waves
constexpr int   kEPL     = kD / 32;      // 20 elements per lane per row
}

typedef __attribute__((ext_vector_type(2))) float v2f;
typedef __attribute__((ext_vector_type(8))) float v8f;

__global__ __launch_bounds__(kThreads)
void dim_masking_fused(const float* __restrict__ x,
                       const float* __restrict__ W1,
                       const float* __restrict__ b1,
                       const float* __restrict__ W2,
                       const float* __restrict__ b2,
                       float* __restrict__ out)
{
    __shared__ float h1s[kRows * kHid];  // 4 KB  : relu(x@W1 + b1)
    __shared__ float hs [kRows * kD];    // 40 KB : h = h1 @ W2 + b2

    const int tid    = threadIdx.x;
    const int lane   = tid & 31;
    const int wave   = tid >> 5;
    const int lanelo = lane & 15;
    const int khalf  = lane >> 4;        // which half of the K/M split
    const int row0   = blockIdx.x * kRows;

    // ---------------- GEMM1: h1 = relu(x_tile @ W1 + b1)  (16 x 64) -------
    // WMMA f32 16x16x4. A frag: a[j] = x[m][k + 2*khalf + j]  (float2 load)
    //                   B frag: b[j] = W1[k + 2*khalf + j][n]
    if (wave < 4) {
        const int    ncol = wave * 16 + lanelo;                  // HID column
        const float* ap   = x  + (size_t)(row0 + lanelo) * kD + 2 * khalf;
        const float* bp   = W1 + (size_t)(2 * khalf) * kHid + ncol;
        v8f acc = {};
        #pragma unroll 8
        for (int k = 0; k < kD; k += 4) {
            v2f a = *(const v2f*)(ap + k);
            const float* w = bp + (size_t)k * kHid;
            v2f b = { w[0], w[kHid] };
            acc = __builtin_amdgcn_wmma_f32_16x16x4_f32(
                false, a, false, b, (short)0, acc, false, false);
        }
        const float bias = b1[ncol];
        #pragma unroll
        for (int r = 0; r < 8; ++r) {
            float v = acc[r] + bias;                              // C: M = r + 8*khalf
            h1s[(r + 8 * khalf) * kHid + ncol] = fmaxf(v, 0.0f);
        }
    }
    __syncthreads();

    // ---------------- GEMM2: h = h1 @ W2 + b2  (16 x 640) -----------------
    #pragma unroll 1
    for (int t = 0; t < 5; ++t) {
        const int    ncol = (wave * 5 + t) * 16 + lanelo;         // D column
        const float* bp   = W2 + (size_t)(2 * khalf) * kD + ncol;
        v8f acc = {};
        #pragma unroll
        for (int k = 0; k < kHid; k += 4) {
            v2f a = *(const v2f*)(&h1s[lanelo * kHid + k + 2 * khalf]);
            const float* w = bp + (size_t)k * kD;
            v2f b = { w[0], w[kD] };
            acc = __builtin_amdgcn_wmma_f32_16x16x4_f32(
                false, a, false, b, (short)0, acc, false, false);
        }
        const float bias = b2[ncol];
        #pragma unroll
        for (int r = 0; r < 8; ++r)
            hs[(r + 8 * khalf) * kD + ncol] = acc[r] + bias;
    }
    __syncthreads();

    // ---------------- iterative soft masking (register-resident) ----------
    // wave owns rows 2*wave and 2*wave+1; lane owns columns  j*32 + lane.
    float hv[2][kEPL], hh[2][kEPL], zb[kEPL];
    #pragma unroll
    for (int s = 0; s < 2; ++s) {
        const int lr = 2 * wave + s;
        #pragma unroll
        for (int j = 0; j < kEPL; ++j) {
            hv[s][j] = hs[lr * kD + j * 32 + lane];
            hh[s][j] = 0.0f;
        }
    }

    #pragma unroll 1
    for (int it = 0; it < kIter; ++it) {
        #pragma unroll
        for (int s = 0; s < 2; ++s) {
            // z = (log(1 - hhat + eps) - h) / TEMP ; row max for stable softmax
            float zmax = -3.402823466e38f;
            #pragma unroll
            for (int j = 0; j < kEPL; ++j) {
                float m = 1.0f - hh[s][j];
                float z = (__logf(m + kEps) - hv[s][j]) * kInvTemp;
                zb[j] = z;
                zmax  = fmaxf(zmax, z);
            }
            #pragma unroll
            for (int off = 16; off > 0; off >>= 1)
                zmax = fmaxf(zmax, __shfl_xor(zmax, off, 32));
            float sum = 0.0f;
            #pragma unroll
            for (int j = 0; j < kEPL; ++j) {
                float e = __expf(zb[j] - zmax);
                zb[j] = e;
                sum  += e;
            }
            #pragma unroll
            for (int off = 16; off > 0; off >>= 1)
                sum += __shfl_xor(sum, off, 32);
            const float inv = 1.0f / sum;
            // hhat += softmax * (1 - hhat)
            #pragma unroll
            for (int j = 0; j < kEPL; ++j) {
                float m = 1.0f - hh[s][j];
                hh[s][j] += zb[j] * inv * m;
            }
        }
    }

    // ---------------- output: (1 - hhat) * x -------------------------------
    #pragma unroll
    for (int s = 0; s < 2; ++s) {
        const size_t grow = (size_t)(row0 + 2 * wave + s) * kD;
        #pragma unroll
        for (int j = 0; j < kEPL; ++j) {
            const size_t idx = grow + (size_t)(j * 32 + lane);
            out[idx] = (1.0f - hh[s][j]) * x[idx];
        }
    }
}

extern "C" void kernel_launch(void* const* d_in, const int* in_sizes, int n_in,
                              void* d_out, int out_size, void* d_ws, size_t ws_size,
                              hipStream_t stream)
{
    const float* x  = (const float*)d_in[0];
    const float* W1 = (const float*)d_in[1];
    const float* b1 = (const float*)d_in[2];
    const float* W2 = (const float*)d_in[3];
    const float* b2 = (const float*)d_in[4];
    float* out = (float*)d_out;

    const int B    = in_sizes[0] / kD;   // 8192 rows
    const int nblk = B / kRows;          // 512 workgroups of 16 rows
    dim_masking_fused<<<nblk, kThreads, 0, stream>>>(x, W1, b1, W2, b2, out);
}